// TraceLevelEncoder_53961969107352
// MI455X (gfx1250) — compile-verified
//
#include <hip/hip_runtime.h>
#include <math.h>

#define NN 50000
#define NE 800000
#define NG 256
#define GRU_H 256

typedef float v2f __attribute__((ext_vector_type(2)));
typedef float v8f __attribute__((ext_vector_type(8)));

// ---------------------------------------------------------------------------
// Generic fp32 WMMA GEMM:  C[M,N] = A[M,K] @ B (+bias)
//   transB==0 : B stored [K,N] row-major  (gcn weights [in,out])
//   transB==1 : B stored [N,K] row-major  (torch [out,in] -> A @ B^T)
// One wave computes a 16x64 strip (4 accumulators); A fragment reused 4x.
// Tile index is wave-uniform, so EXEC is all-ones at every WMMA.
// ---------------------------------------------------------------------------
__global__ __launch_bounds__(256) void gemm_wmma_f32(
    const float* __restrict__ A, const float* __restrict__ B,
    const float* __restrict__ bias, float* __restrict__ C,
    int M, int N, int K, int transB)
{
  const int lane   = threadIdx.x & 31;
  const int wave   = threadIdx.x >> 5;
  const int tilesN = N >> 6;                       // 64-wide strips
  const int tile   = blockIdx.x * 8 + wave;
  const int nTiles = (M >> 4) * tilesN;
  if (tile >= nTiles) return;                      // wave-uniform exit
  const int tm = (tile / tilesN) << 4;
  const int tn = (tile % tilesN) << 6;
  const int r  = lane & 15;                        // row (A) / col (B,C)
  const int kh = lane >> 4;                        // k-pair select {0,1} vs {2,3}

  v8f c0 = {}; v8f c1 = {}; v8f c2 = {}; v8f c3 = {};
  const float* Arow = A + (size_t)(tm + r) * K;

  for (int k0 = 0; k0 < K; k0 += 4) {
    const int kb = k0 + 2 * kh;
    v2f a; a.x = Arow[kb]; a.y = Arow[kb + 1];
    v2f b0, b1, b2, b3;
    if (transB) {
      const float* Bp = B + (size_t)(tn + r) * K + kb;
      b0.x = Bp[0];          b0.y = Bp[1];
      b1.x = Bp[16 * K];     b1.y = Bp[16 * K + 1];
      b2.x = Bp[32 * K];     b2.y = Bp[32 * K + 1];
      b3.x = Bp[48 * K];     b3.y = Bp[48 * K + 1];
    } else {
      const float* Bp = B + (size_t)kb * N + tn + r;
      b0.x = Bp[0];   b0.y = Bp[N];
      b1.x = Bp[16];  b1.y = Bp[16 + N];
      b2.x = Bp[32];  b2.y = Bp[32 + N];
      b3.x = Bp[48];  b3.y = Bp[48 + N];
    }
    c0 = __builtin_amdgcn_wmma_f32_16x16x4_f32(false, a, false, b0, (short)0, c0, false, false);
    c1 = __builtin_amdgcn_wmma_f32_16x16x4_f32(false, a, false, b1, (short)0, c1, false, false);
    c2 = __builtin_amdgcn_wmma_f32_16x16x4_f32(false, a, false, b2, (short)0, c2, false, false);
    c3 = __builtin_amdgcn_wmma_f32_16x16x4_f32(false, a, false, b3, (short)0, c3, false, false);
  }

  float bs0 = 0.f, bs1 = 0.f, bs2 = 0.f, bs3 = 0.f;
  if (bias) {
    bs0 = bias[tn + r];      bs1 = bias[tn + 16 + r];
    bs2 = bias[tn + 32 + r]; bs3 = bias[tn + 48 + r];
  }
#pragma unroll
  for (int j = 0; j < 8; ++j) {                    // VGPR j -> rows j / j+8
    const int row = tm + j + 8 * kh;
    float* Cr = C + (size_t)row * N + tn + r;
    Cr[0]  = c0[j] + bs0;
    Cr[16] = c1[j] + bs1;
    Cr[32] = c2[j] + bs2;
    Cr[48] = c3[j] + bs3;
  }
}

// ---------------------------------------------------------------------------
// GCN normalization helpers
// ---------------------------------------------------------------------------
__global__ void init_deg(float* deg) {
  int i = blockIdx.x * blockDim.x + threadIdx.x;
  if (i < NN) deg[i] = 1.0f;                       // self-loop
}
__global__ void edge_deg(const int* __restrict__ dst, float* deg) {
  int e = blockIdx.x * blockDim.x + threadIdx.x;
  if (e < NE) atomicAdd(&deg[dst[e]], 1.0f);
}
__global__ void deg_to_dinv(float* deg) {
  int i = blockIdx.x * blockDim.x + threadIdx.x;
  if (i < NN) deg[i] = rsqrtf(deg[i]);             // deg >= 1 always
}

// ---------------------------------------------------------------------------
// Aggregation: out = relu( sum_{e: dst=i} norm_e * H[src_e] + dinv_i^2*H[i] + b )
// ---------------------------------------------------------------------------
__global__ void agg_init(const float* __restrict__ H, const float* __restrict__ dinv,
                         float* __restrict__ out, int total, int dshift) {
  int t = blockIdx.x * blockDim.x + threadIdx.x;
  if (t >= total) return;
  const int i = t >> dshift;
  const float di = dinv[i];
  out[t] = H[t] * di * di;                          // self-loop term, also zero-inits
}

__global__ void agg_edges(const float* __restrict__ H, const float* __restrict__ dinv,
                          const int* __restrict__ src, const int* __restrict__ dst,
                          float* __restrict__ out, int dout, int cshift, int total) {
  int t = blockIdx.x * blockDim.x + threadIdx.x;
  if (t >= total) return;
  const int chunks = 1 << cshift;
  const int e = t >> cshift;
  const int c = t & (chunks - 1);
  const int s = src[e], d = dst[e];
  const float w = dinv[s] * dinv[d];
  const float4 h4 = ((const float4*)(H + (size_t)s * dout))[c];
  float* o = out + (size_t)d * dout + c * 4;
  atomicAdd(o + 0, h4.x * w);
  atomicAdd(o + 1, h4.y * w);
  atomicAdd(o + 2, h4.z * w);
  atomicAdd(o + 3, h4.w * w);
}

__global__ void agg_finish(float* __restrict__ x, const float* __restrict__ b,
                           int total, int dmask) {
  int t = blockIdx.x * blockDim.x + threadIdx.x;
  if (t >= total) return;
  x[t] = fmaxf(x[t] + b[t & dmask], 0.0f);
}

// ---------------------------------------------------------------------------
// Attention pooling
// ---------------------------------------------------------------------------
__global__ __launch_bounds__(256) void gate_dot(const float* __restrict__ x,
                                                const float* __restrict__ gw,
                                                const float* __restrict__ gb,
                                                float* __restrict__ gate) {
  const int lane = threadIdx.x & 31;
  const int node = (blockIdx.x * 8) + (threadIdx.x >> 5);
  if (node >= NN) return;
  const float* row = x + (size_t)node * 512;
  float s = 0.f;
#pragma unroll
  for (int k = 0; k < 16; ++k) s += row[lane + 32 * k] * gw[lane + 32 * k];
#pragma unroll
  for (int off = 16; off > 0; off >>= 1) s += __shfl_down(s, off, 32);
  if (lane == 0) gate[node] = s + gb[0];
}

__global__ void graphs_init(float* __restrict__ emb, float* __restrict__ gmax,
                            float* __restrict__ denom) {
  int t = blockIdx.x * blockDim.x + threadIdx.x;
  if (t < NG * 512) emb[t] = 0.f;
  if (t < NG) {
    ((int*)gmax)[t] = 0xFF800000;                   // -inf
    denom[t] = 0.f;
  }
}

__device__ inline void atomicMaxF(float* addr, float v) {
  if (v >= 0.f) atomicMax((int*)addr, __float_as_int(v));
  else          atomicMin((unsigned int*)addr, (unsigned int)__float_as_int(v));
}

__global__ void gate_max(const float* __restrict__ gate, const int* __restrict__ batch,
                         float* __restrict__ gmax) {
  int i = blockIdx.x * blockDim.x + threadIdx.x;
  if (i < NN) atomicMaxF(&gmax[batch[i]], gate[i]);
}

__global__ void gate_exp(float* __restrict__ gate, const int* __restrict__ batch,
                         const float* __restrict__ gmax, float* __restrict__ denom) {
  int i = blockIdx.x * blockDim.x + threadIdx.x;
  if (i >= NN) return;
  const float e = expf(gate[i] - gmax[batch[i]]);
  gate[i] = e;
  atomicAdd(&denom[batch[i]], e);
}

__global__ void emb_scatter(const float* __restrict__ x, const float* __restrict__ gate,
                            const int* __restrict__ batch, const float* __restrict__ denom,
                            float* __restrict__ emb, int total) {
  int t = blockIdx.x * blockDim.x + threadIdx.x;
  if (t >= total) return;
  const int i = t >> 7;                             // 128 float4-chunks of 512
  const int c = t & 127;
  const int g = batch[i];
  const float alpha = gate[i] / denom[g];
  const float4 x4 = ((const float4*)(x + (size_t)i * 512))[c];
  float* o = emb + (size_t)g * 512 + c * 4;
  atomicAdd(o + 0, x4.x * alpha);
  atomicAdd(o + 1, x4.y * alpha);
  atomicAdd(o + 2, x4.z * alpha);
  atomicAdd(o + 3, x4.w * alpha);
}

// ---------------------------------------------------------------------------
// GRU cell with h=0:  gh = bhh ;  h' = (1-z) * n
// gi already contains x@Wih^T + bih (from the WMMA GEMM).
// ---------------------------------------------------------------------------
__global__ void gru_elem(const float* __restrict__ gi, const float* __restrict__ bhh,
                         float* __restrict__ hout) {
  int t = blockIdx.x * blockDim.x + threadIdx.x;
  if (t >= NG * GRU_H) return;
  const int g = t >> 8;
  const int j = t & 255;
  const float ir = gi[(size_t)g * 768 + j];
  const float iz = gi[(size_t)g * 768 + 256 + j];
  const float in = gi[(size_t)g * 768 + 512 + j];
  const float rr = 1.f / (1.f + expf(-(ir + bhh[j])));
  const float zz = 1.f / (1.f + expf(-(iz + bhh[256 + j])));
  const float nn = tanhf(in + rr * bhh[512 + j]);
  hout[t] = (1.f - zz) * nn;
}

// ---------------------------------------------------------------------------
// Launch
// ---------------------------------------------------------------------------
extern "C" void kernel_launch(void* const* d_in, const int* in_sizes, int n_in,
                              void* d_out, int out_size, void* d_ws, size_t ws_size,
                              hipStream_t stream) {
  const float* x0      = (const float*)d_in[0];
  const int*   eidx    = (const int*)d_in[1];     // [2, NE]: row0=src, row1=dst
  const int*   batch   = (const int*)d_in[2];
  const float* gcn_w[3] = { (const float*)d_in[3], (const float*)d_in[5], (const float*)d_in[7] };
  const float* gcn_b[3] = { (const float*)d_in[4], (const float*)d_in[6], (const float*)d_in[8] };
  const float* gate_w  = (const float*)d_in[9];
  const float* gate_b  = (const float*)d_in[10];
  const float* wih0 = (const float*)d_in[11];
  const float* bih0 = (const float*)d_in[13];
  const float* bhh0 = (const float*)d_in[14];
  const float* wih1 = (const float*)d_in[15];
  const float* bih1 = (const float*)d_in[17];
  const float* bhh1 = (const float*)d_in[18];
  const float* proj_w = (const float*)d_in[19];
  const float* proj_b = (const float*)d_in[20];
  float* out = (float*)d_out;

  const int* src = eidx;
  const int* dst = eidx + NE;

  float* w = (float*)d_ws;
  float* dinv  = w;                 // 50048
  float* H     = dinv + 50048;      // 25.6M
  float* X     = H + 25600000;      // 25.6M
  float* gate  = X + 25600000;      // 50048
  float* gmax  = gate + 50048;      // 256
  float* denom = gmax + 256;        // 256
  float* emb   = denom + 256;       // 131072
  float* gi    = emb + 131072;      // 196608
  float* h1    = gi + 196608;       // 65536
  float* h2    = h1 + 65536;        // 65536

  const int B = 256;
  auto cdiv = [](long long a, long long b) { return (int)((a + b - 1) / b); };

  // --- degree / normalization ---
  init_deg<<<cdiv(NN, B), B, 0, stream>>>(dinv);
  edge_deg<<<cdiv(NE, B), B, 0, stream>>>(dst, dinv);
  deg_to_dinv<<<cdiv(NN, B), B, 0, stream>>>(dinv);

  // --- 3 GCN layers ---
  const int dims[4] = {64, 128, 256, 512};
  const float* xin = x0;
  for (int l = 0; l < 3; ++l) {
    const int K = dims[l], N = dims[l + 1];
    const int tiles = (NN / 16) * (N / 64);
    gemm_wmma_f32<<<cdiv(tiles, 8), B, 0, stream>>>(xin, gcn_w[l], nullptr, H, NN, N, K, 0);

    int dshift = (N == 128) ? 7 : (N == 256) ? 8 : 9;
    int total = NN * N;
    agg_init<<<cdiv(total, B), B, 0, stream>>>(H, dinv, X, total, dshift);

    int cshift = dshift - 2;                        // float4 chunks = N/4
    int etotal = NE * (N / 4);
    agg_edges<<<cdiv(etotal, B), B, 0, stream>>>(H, dinv, src, dst, X, N, cshift, etotal);

    agg_finish<<<cdiv(total, B), B, 0, stream>>>(X, gcn_b[l], total, N - 1);
    xin = X;
  }

  // --- attention pooling ---
  gate_dot<<<cdiv(NN, 8), B, 0, stream>>>(X, gate_w, gate_b, gate);
  graphs_init<<<cdiv(NG * 512, B), B, 0, stream>>>(emb, gmax, denom);
  gate_max<<<cdiv(NN, B), B, 0, stream>>>(gate, batch, gmax);
  gate_exp<<<cdiv(NN, B), B, 0, stream>>>(gate, batch, gmax, denom);
  emb_scatter<<<cdiv(NN * 128, B), B, 0, stream>>>(X, gate, batch, denom, emb, NN * 128);

  // --- GRU layer 0: gi = emb @ wih0^T + bih0 ; h1 = (1-z)*n ---
  {
    const int tiles = (NG / 16) * (768 / 64);
    gemm_wmma_f32<<<cdiv(tiles, 8), B, 0, stream>>>(emb, wih0, bih0, gi, NG, 768, 512, 1);
    gru_elem<<<cdiv(NG * GRU_H, B), B, 0, stream>>>(gi, bhh0, h1);
  }
  // --- GRU layer 1 (hidden is h0 = 0, input is h1) ---
  {
    const int tiles = (NG / 16) * (768 / 64);
    gemm_wmma_f32<<<cdiv(tiles, 8), B, 0, stream>>>(h1, wih1, bih1, gi, NG, 768, 256, 1);
    gru_elem<<<cdiv(NG * GRU_H, B), B, 0, stream>>>(gi, bhh1, h2);
  }
  // --- output projection: out = h2 @ proj_w^T + proj_b ---
  {
    const int tiles = (NG / 16) * (512 / 64);
    gemm_wmma_f32<<<cdiv(tiles, 8), B, 0, stream>>>(h2, proj_w, proj_b, out, NG, 512, 256, 1);
  }
}